// GraphAttentionLayer_23381801959470
// MI455X (gfx1250) — compile-verified
//
#include <hip/hip_runtime.h>
#include <hip/hip_bf16.h>
#include <math.h>

// Problem constants (reference: B=8, N=2048, Fin=256, Fout=128, f32)
#define B_   8
#define N_   2048
#define FIN  256
#define FOUT 128
#define BN   (B_ * N_)
#define ALPHA 0.2f

typedef float v2f __attribute__((ext_vector_type(2)));
typedef float v8f __attribute__((ext_vector_type(8)));
typedef unsigned int u32x4 __attribute__((ext_vector_type(4)));
typedef int i32x8 __attribute__((ext_vector_type(8)));
typedef int i32x4 __attribute__((ext_vector_type(4)));

__device__ __forceinline__ float leaky(float x) {
    // x>=0 -> x ; x<0 -> 0.2x  == max(x, 0.2x)
    return fmaxf(x, ALPHA * x);
}

// ---------------------------------------------------------------------------
// K1: Wh = h @ W   (BN x FIN) @ (FIN x FOUT) with V_WMMA_F32_16X16X4_F32.
// One wave owns a 16-row tile and ALL 8 column tiles (Fout=128 = 8*16).
// ---------------------------------------------------------------------------
__global__ __launch_bounds__(256) void gemm1_wmma(const float* __restrict__ h,
                                                  const float* __restrict__ W,
                                                  float* __restrict__ Wh) {
    const int lane  = threadIdx.x & 31;
    const int wave  = threadIdx.x >> 5;
    const int rt    = blockIdx.x * 8 + wave;   // row tile 0..BN/16-1
    const int row0  = rt * 16;
    const int mn    = lane & 15;               // A: row M ; B/D: col N
    const int khalf = (lane >> 4) << 1;        // lanes 0-15 -> K{0,1}, 16-31 -> K{2,3}

    v8f zero = {};
    v8f acc[8];
#pragma unroll
    for (int t = 0; t < 8; ++t) acc[t] = zero;

    const float* hrow = h + (size_t)(row0 + mn) * FIN;

    for (int ks = 0; ks < FIN; ks += 4) {
        v2f a;
        a.x = hrow[ks + khalf];
        a.y = hrow[ks + khalf + 1];
        const float* w0 = W + (size_t)(ks + khalf) * FOUT + mn;
        const float* w1 = w0 + FOUT;
#pragma unroll
        for (int t = 0; t < 8; ++t) {
            v2f b;
            b.x = w0[t * 16];
            b.y = w1[t * 16];
            acc[t] = __builtin_amdgcn_wmma_f32_16x16x4_f32(
                false, a, false, b, (short)0, acc[t], false, false);
        }
    }

    const int rbase = row0 + ((lane >> 4) << 3);   // D: VGPR r -> M = r + 8*(lane>=16)
#pragma unroll
    for (int t = 0; t < 8; ++t)
#pragma unroll
        for (int r = 0; r < 8; ++r)
            Wh[(size_t)(rbase + r) * FOUT + mn + t * 16] = acc[t][r];
}

// ---------------------------------------------------------------------------
// K2: s1[row] = Wh[row,:].a1 ; s2[row] = Wh[row,:].a2 (wave per row)
// ---------------------------------------------------------------------------
__global__ __launch_bounds__(256) void row_dots(const float* __restrict__ Wh,
                                                const float* __restrict__ a,
                                                float* __restrict__ s1,
                                                float* __restrict__ s2) {
    const int lane = threadIdx.x & 31;
    const int row  = blockIdx.x * 8 + (threadIdx.x >> 5);
    const float* wr = Wh + (size_t)row * FOUT + lane * 4;
    float d1 = 0.f, d2 = 0.f;
#pragma unroll
    for (int q = 0; q < 4; ++q) {
        float v = wr[q];
        d1 += v * a[lane * 4 + q];
        d2 += v * a[FOUT + lane * 4 + q];
    }
#pragma unroll
    for (int off = 16; off > 0; off >>= 1) {
        d1 += __shfl_xor(d1, off, 32);
        d2 += __shfl_xor(d2, off, 32);
    }
    if (lane == 0) { s1[row] = d1; s2[row] = d2; }
}

// ---------------------------------------------------------------------------
// K3: maxS1[b] = max_i s1[b,i]  (one block per batch)
// ---------------------------------------------------------------------------
__global__ __launch_bounds__(256) void batch_max(const float* __restrict__ s1,
                                                 float* __restrict__ maxS1) {
    __shared__ float sm[256];
    const int b = blockIdx.x;
    float m = -3.4e38f;
    for (int i = threadIdx.x; i < N_; i += 256) m = fmaxf(m, s1[b * N_ + i]);
    sm[threadIdx.x] = m;
    __syncthreads();
    for (int s = 128; s > 0; s >>= 1) {
        if ((int)threadIdx.x < s) sm[threadIdx.x] = fmaxf(sm[threadIdx.x], sm[threadIdx.x + s]);
        __syncthreads();
    }
    if (threadIdx.x == 0) maxS1[b] = sm[0];
}

// ---------------------------------------------------------------------------
// K4: denom[b,j] = sum_i exp(leaky(s1_i+s2_j) - m_j), m_j = leaky(maxS1+s2_j)
//     then scale Wh[b,j,:] *= 1/denom in place (Wh becomes V). Wave per (b,j).
// ---------------------------------------------------------------------------
__global__ __launch_bounds__(256) void col_norm(float* __restrict__ Wh,
                                                const float* __restrict__ s1,
                                                const float* __restrict__ s2,
                                                const float* __restrict__ maxS1) {
    const int lane = threadIdx.x & 31;
    const int gj   = blockIdx.x * 8 + (threadIdx.x >> 5);  // 0..BN-1
    const int b    = gj >> 11;                             // /N_
    const float s2j = s2[gj];
    const float mj  = leaky(maxS1[b] + s2j);
    const float* s1b = s1 + b * N_;
    float sum = 0.f;
    for (int i = lane; i < N_; i += 32)
        sum += __expf(leaky(s1b[i] + s2j) - mj);
#pragma unroll
    for (int off = 16; off > 0; off >>= 1) sum += __shfl_xor(sum, off, 32);
    const float inv = 1.0f / sum;                          // all lanes hold full sum
    float* row = Wh + (size_t)gj * FOUT;
#pragma unroll
    for (int q = 0; q < 4; ++q) row[lane + 32 * q] *= inv;
}

// ---------------------------------------------------------------------------
// TDM helper: issue tensor_load_to_lds of a 32-row x 128-col f32 tile of V[b]
// starting at global address `ga` into LDS byte address `la`.
// D# group1 packing (see CDNA5 ISA), data_size=4B:
//  w0 = data_size(2)<<16                      = 0x00020000
//  w1 = tensor_dim0[15:0]  (=128) << 16       = 0x00800000
//  w2 = tensor_dim1[15:0]  (=2048) << 16      = 0x08000000
//  w3 = tile_dim0          (=128) << 16       = 0x00800000
//  w4 = tile_dim1          (=32)              = 32
//  w5 = tensor_dim0_stride (=128)             = 128
//  w6 = w7 = 0
// ---------------------------------------------------------------------------
__device__ __forceinline__ void tdm_load_tile(unsigned long long ga, unsigned int la) {
    const i32x8 g1 = { 0x00020000, 0x00800000, 0x08000000, 0x00800000, 32, 128, 0, 0 };
    const i32x4 gz4 = {};
    const i32x8 gz8 = {};
    // group0: count=1 | lds_addr | global_addr[56:0] | type=2 (bits 127:126)
    u32x4 g0;
    g0.x = 1u;
    g0.y = la;
    g0.z = (unsigned int)ga;
    g0.w = (unsigned int)((ga >> 32) & 0x01FFFFFFull) | 0x80000000u;
    // 6-arg form (amdgpu-toolchain / clang-23): (g0, g1, g2, g3, g4, cpol)
    __builtin_amdgcn_tensor_load_to_lds(g0, g1, gz4, gz4, gz8, 0);
}

// ---------------------------------------------------------------------------
// K5: out[b,i,:] = elu( sum_j P[b,i,j] * V[b,j,:] ),
//     P = exp(leaky(s1_i+s2_j) - m_j) generated in registers.
//     Block = 8 waves = 8 row tiles (128 i rows) of one batch.
//     V K-blocks (32 j rows) staged into LDS by the Tensor Data Mover,
//     double buffered; all 8 waves share the tile (8x L2-traffic cut).
// ---------------------------------------------------------------------------
__global__ __launch_bounds__(256) void gemm2_wmma(const float* __restrict__ V,
                                                  const float* __restrict__ s1,
                                                  const float* __restrict__ s2,
                                                  const float* __restrict__ maxS1,
                                                  float* __restrict__ out) {
    __shared__ float lds[2][32 * FOUT];        // 2 x 16 KB double buffer

    const int lane  = threadIdx.x & 31;
    const int wave  = threadIdx.x >> 5;
    const int rt    = blockIdx.x * 8 + wave;   // 0..BN/16-1
    const int b     = rt >> 7;                 // 128 row tiles per batch
    const int i0    = (rt & 127) * 16;
    const int mn    = lane & 15;
    const int khalf = (lane >> 4) << 1;

    const float s1v  = s1[b * N_ + i0 + mn];   // A row for this lane
    const float maxb = maxS1[b];
    const float* s2b = s2 + b * N_;
    const float* Vb  = V + (size_t)b * N_ * FOUT;

    v8f zero = {};
    v8f acc[8];
#pragma unroll
    for (int t = 0; t < 8; ++t) acc[t] = zero;

    // prologue: stage K-block 0 (one TDM issue per workgroup, by wave 0)
    if (wave == 0) {
        tdm_load_tile((unsigned long long)(size_t)Vb,
                      (unsigned int)(size_t)&lds[0][0]);
        __builtin_amdgcn_s_wait_tensorcnt(0);
    }
    __syncthreads();

    for (int kb = 0; kb < N_; kb += 32) {
        const int cur = (kb >> 5) & 1;
        if (wave == 0 && kb + 32 < N_) {       // prefetch next K-block via TDM
            tdm_load_tile((unsigned long long)(size_t)(Vb + (size_t)(kb + 32) * FOUT),
                          (unsigned int)(size_t)&lds[cur ^ 1][0]);
        }

        const float* lbuf = &lds[cur][0];
#pragma unroll
        for (int g = 0; g < 8; ++g) {          // 8 K-steps of 4 within the block
            const int j0 = kb + 4 * g + khalf;
            const float s20 = s2b[j0];
            const float s21 = s2b[j0 + 1];
            v2f a;
            a.x = __expf(leaky(s1v + s20) - leaky(maxb + s20));
            a.y = __expf(leaky(s1v + s21) - leaky(maxb + s21));

            const float* lb = lbuf + (4 * g + khalf) * FOUT + mn;
#pragma unroll
            for (int t = 0; t < 8; ++t) {
                v2f bb;
                bb.x = lb[t * 16];             // row j0
                bb.y = lb[t * 16 + FOUT];      // row j0+1 (+512B -> ds_load_2addr)
                acc[t] = __builtin_amdgcn_wmma_f32_16x16x4_f32(
                    false, a, false, bb, (short)0, acc[t], false, false);
            }
        }

        if (wave == 0) __builtin_amdgcn_s_wait_tensorcnt(0);
        __syncthreads();                        // next buffer ready, reads done
    }

    const int rbase = i0 + ((lane >> 4) << 3);
#pragma unroll
    for (int t = 0; t < 8; ++t)
#pragma unroll
        for (int r = 0; r < 8; ++r) {
            float x = acc[t][r];
            x = (x > 0.f) ? x : (__expf(x) - 1.0f);   // ELU(alpha=1)
            out[((size_t)b * N_ + rbase + r) * FOUT + mn + t * 16] = x;
        }
}

// ---------------------------------------------------------------------------
extern "C" void kernel_launch(void* const* d_in, const int* in_sizes, int n_in,
                              void* d_out, int out_size, void* d_ws, size_t ws_size,
                              hipStream_t stream) {
    const float* h = (const float*)d_in[0];   // (8,2048,256)
    const float* W = (const float*)d_in[1];   // (256,128)
    const float* a = (const float*)d_in[2];   // (256,1)
    float* out = (float*)d_out;               // (8,2048,128)

    // workspace layout (floats)
    float* wsf   = (float*)d_ws;
    float* Wh    = wsf;                        // BN*FOUT  (becomes V in place)
    float* s1    = Wh + (size_t)BN * FOUT;     // BN
    float* s2    = s1 + BN;                    // BN
    float* maxS1 = s2 + BN;                    // B_

    // K1: Wh = h @ W           (BN/16 row tiles, 8 waves per block)
    gemm1_wmma<<<BN / 16 / 8, 256, 0, stream>>>(h, W, Wh);
    // K2: s1,s2 row dots       (wave per row)
    row_dots<<<BN / 8, 256, 0, stream>>>(Wh, a, s1, s2);
    // K3: per-batch max of s1
    batch_max<<<B_, 256, 0, stream>>>(s1, maxS1);
    // K4: softmax denominators + fold into Wh -> V (in place)
    col_norm<<<BN / 8, 256, 0, stream>>>(Wh, s1, s2, maxS1);
    // K5: out = elu(P @ V), V staged through LDS by the Tensor Data Mover
    gemm2_wmma<<<BN / 16 / 8, 256, 0, stream>>>(Wh, s1, s2, maxS1, out);
}